// BakeAugment_25520695673132
// MI455X (gfx1250) — compile-verified
//
#include <hip/hip_runtime.h>

typedef float v2f __attribute__((ext_vector_type(2)));
typedef float v8f __attribute__((ext_vector_type(8)));

#define NPIX  12582912   // 16*3*512*512
#define PLANE 262144     // 512*512

// Orthonormal 8x8 DCT-II matrix (rows = frequency k, cols = n)
__constant__ float c_dct[64] = {
  0.35355339f, 0.35355339f, 0.35355339f, 0.35355339f, 0.35355339f, 0.35355339f, 0.35355339f, 0.35355339f,
  0.49039264f, 0.41573481f, 0.27778512f, 0.09754516f,-0.09754516f,-0.27778512f,-0.41573481f,-0.49039264f,
  0.46193977f, 0.19134172f,-0.19134172f,-0.46193977f,-0.46193977f,-0.19134172f, 0.19134172f, 0.46193977f,
  0.41573481f,-0.09754516f,-0.49039264f,-0.27778512f, 0.27778512f, 0.49039264f, 0.09754516f,-0.41573481f,
  0.35355339f,-0.35355339f,-0.35355339f, 0.35355339f, 0.35355339f,-0.35355339f,-0.35355339f, 0.35355339f,
  0.27778512f,-0.49039264f, 0.09754516f, 0.41573481f,-0.41573481f,-0.09754516f, 0.49039264f,-0.27778512f,
  0.19134172f,-0.46193977f, 0.46193977f,-0.19134172f,-0.19134172f, 0.46193977f,-0.46193977f, 0.19134172f,
  0.09754516f,-0.27778512f, 0.41573481f,-0.49039264f, 0.49039264f,-0.41573481f, 0.27778512f,-0.09754516f,
};

__constant__ float c_q[64] = {
  16,11,10,16,24,40,51,61,
  12,12,14,19,26,58,60,55,
  14,13,16,24,40,57,69,56,
  14,17,22,29,51,87,80,62,
  18,22,37,56,68,109,103,77,
  24,35,55,64,81,104,113,92,
  49,64,78,87,103,121,120,101,
  72,92,95,98,112,100,103,99,
};

__device__ __forceinline__ float clip01(float v) { return fminf(fmaxf(v, 0.0f), 1.0f); }

// x^0.9 for x in [1e-8, 1]: raw v_log_f32 / v_exp_f32 (base-2), 3 VALU ops.
__device__ __forceinline__ float pow09(float x) {
  return __builtin_amdgcn_exp2f(0.9f * __builtin_amdgcn_logf(x));
}

// lane <-> lane^16 exchange: v_permlanex16 with identity selectors (no LDS pipe).
__device__ __forceinline__ float xhalf(float v) {
  int i = __builtin_bit_cast(int, v);
  i = __builtin_amdgcn_permlanex16(i, i, 0x76543210, 0xfedcba98, false, false);
  return __builtin_bit_cast(float, i);
}

// lane <-> lane^1 exchange: v_mov_b32_dpp8 with selector (1,0,3,2,5,4,7,6).
__device__ __forceinline__ float xone(float v) {
  int i = __builtin_bit_cast(int, v);
  i = __builtin_amdgcn_mov_dpp8(i, 0x00DE54C1);
  return __builtin_bit_cast(float, i);
}

// Convert a 16x16 matrix held in WMMA C/D layout (xr[v] = X[v+8*hf, col])
// into the j-th B-operand chunk (rows 4j..4j+3 of X) for f32 16x16x4 WMMA.
__device__ __forceinline__ v2f bchunkJ(const float xr[8], int j, int hf) {
  const int p = 4 * (j & 1);
  v2f r;
  if (j < 2) {  // rows 4j..4j+3 live in lanes 0-15 (half 0)
    float s0 = xhalf(xr[p + 2]);
    float s1 = xhalf(xr[p + 3]);
    r.x = hf ? s0 : xr[p];
    r.y = hf ? s1 : xr[p + 1];
  } else {      // rows live in lanes 16-31 (half 1)
    float s0 = xhalf(xr[p]);
    float s1 = xhalf(xr[p + 1]);
    r.x = hf ? xr[p + 2] : s0;
    r.y = hf ? xr[p + 3] : s1;
  }
  return r;
}

// D(16x16) = A(16x16) * B(16x16) as 4 chained K=4 f32 WMMAs.
__device__ __forceinline__ void wmma4(const v2f a[4], const v2f b[4], float out[8]) {
  v8f acc = {};
#pragma unroll
  for (int j = 0; j < 4; ++j)
    acc = __builtin_amdgcn_wmma_f32_16x16x4_f32(false, a[j], false, b[j],
                                                (short)0, acc, false, false);
#pragma unroll
  for (int v = 0; v < 8; ++v) out[v] = acc[v];
}

// Store C-layout tile to LDS (row-major, stride 20 -> bank-conflict-free),
// reload B-layout chunks of the TRANSPOSE as b64. Wave-local; LDS in-order.
__device__ __forceinline__ void lds_transpose_b(float* l, const float in[8],
                                                int col, int hf, v2f b[4]) {
#pragma unroll
  for (int v = 0; v < 8; ++v) l[(v + 8 * hf) * 20 + col] = in[v];
#pragma unroll
  for (int j = 0; j < 4; ++j) {
    // byte address = 4*(col*20 + 4j + 2hf): multiple of 8 -> aligned b64 load
    b[j] = *reinterpret_cast<const v2f*>(&l[col * 20 + 4 * j + 2 * hf]);
  }
}

__global__ __launch_bounds__(256) void bake_aug_kernel(
    const float* __restrict__ xin, const float* __restrict__ dither,
    const float* __restrict__ gauss, const float* __restrict__ shift,
    float* __restrict__ out) {
  __shared__ float lds[8][320];  // 16 rows * stride 20, per wave
  const int lane = threadIdx.x & 31;
  const int wv   = threadIdx.x >> 5;
  const int hf   = lane >> 4;   // which half of the wave
  const int col  = lane & 15;   // N index (x within tile)

  const int tile = blockIdx.x * 8 + wv;        // 16384 tiles total
  const int b    = tile >> 10;
  const int tr   = tile & 1023;
  const int y0   = (tr >> 5) << 4;
  const int xc   = ((tr & 31) << 4) + col;

  // A operands: block-diagonal D16 and D16^T in 16x4-chunk A-layout.
  v2f aD[4], aDt[4];
#pragma unroll
  for (int j = 0; j < 4; ++j) {
    const int k = 4 * j + 2 * hf;
    const bool own = (col >> 3) == (j >> 1);   // same 8x8 diagonal block?
    aD[j].x  = own ? c_dct[(col & 7) * 8 + (k & 7)]       : 0.0f;
    aD[j].y  = own ? c_dct[(col & 7) * 8 + ((k + 1) & 7)] : 0.0f;
    aDt[j].x = own ? c_dct[(k & 7) * 8 + (col & 7)]       : 0.0f;
    aDt[j].y = own ? c_dct[((k + 1) & 7) * 8 + (col & 7)] : 0.0f;
  }

  // Quant factors for this lane's 8 elements of Y^T (i = y%8 = lane&7, j = v),
  // plus reciprocals (one-time v_rcp instead of 24 in-loop divides).
  float q[8], qinv[8];
#pragma unroll
  for (int v = 0; v < 8; ++v) {
    q[v] = c_q[(lane & 7) * 8 + v] * (50.0f / 15.0f) + 1e-5f;
    qinv[v] = __builtin_amdgcn_rcpf(q[v]);
  }

  const float sh[3] = { shift[0] * 0.05f, shift[1] * 0.05f, shift[2] * 0.05f };
  // Fold the hf-dependent row offset into the base so every global access is
  // base + compile-time-constant immediate offset.
  const int base = (b * 3) * PLANE + (y0 + 8 * hf) * 512 + xc;
  const float* __restrict__ xin_b  = xin    + base;
  const float* __restrict__ dit_b  = dither + base;
  const float* __restrict__ gau_b  = gauss  + base;
  float* __restrict__ outI = out + base;
  float* __restrict__ outT = out + (NPIX + base);

  // Load, emit target path, dither-quantize, RGB->YCbCr
  float ycc[3][8];
  {
    float rgbq[3][8];
#pragma unroll
    for (int c = 0; c < 3; ++c) {
#pragma unroll
      for (int v = 0; v < 8; ++v) {
        const int off = c * PLANE + v * 512;   // constant per unrolled iter
        const float xv = __builtin_nontemporal_load(xin_b + off);
        const float dv = __builtin_nontemporal_load(dit_b + off);
        float tg = xv + sh[c];
        tg = fminf(fmaxf(tg, 1e-8f), 1.0f);
        __builtin_nontemporal_store(clip01(pow09(tg)), outT + off);
        const float iv = xv + (dv - 0.5f) * (1.0f / 15.0f);
        rgbq[c][v] = rintf(iv * 15.0f) * (1.0f / 15.0f);  // RNE = jnp.round
      }
    }
#pragma unroll
    for (int v = 0; v < 8; ++v) {
      const float r = rgbq[0][v], g = rgbq[1][v], bl = rgbq[2][v];
      ycc[0][v] =  0.299f  * r + 0.587f  * g + 0.114f  * bl;
      ycc[1][v] = -0.1687f * r - 0.3313f * g + 0.5f    * bl + 0.5f;
      ycc[2][v] =  0.5f    * r - 0.4187f * g - 0.0813f * bl + 0.5f;
    }
  }

  // Fused chroma 2x2 downsample + nearest upsample: in-register 2x2 mean.
#pragma unroll
  for (int c = 1; c < 3; ++c) {
#pragma unroll
    for (int v = 0; v < 8; v += 2) {
      float s = ycc[c][v] + ycc[c][v + 1];    // row pair (within lane)
      s += xone(s);                           // col pair (adjacent lane, dpp8)
      s *= 0.25f;
      ycc[c][v] = s; ycc[c][v + 1] = s;
    }
  }

  // Per-channel 2D DCT -> quantize -> inverse 2D DCT (4 WMMA stages each).
  float rec[3][8];
#pragma unroll
  for (int c = 0; c < 3; ++c) {
    v2f bb[4];
#pragma unroll
    for (int j = 0; j < 4; ++j) bb[j] = bchunkJ(ycc[c], j, hf);
    float t1[8];
    wmma4(aD, bb, t1);                         // T = D16 * P
    lds_transpose_b(lds[wv], t1, col, hf, bb);
    float t2[8];
    wmma4(aD, bb, t2);                         // Y^T = D16 * T^T = (T*D16^T)^T
#pragma unroll
    for (int v = 0; v < 8; ++v) t2[v] = rintf(t2[v] * qinv[v]) * q[v];
#pragma unroll
    for (int j = 0; j < 4; ++j) bb[j] = bchunkJ(t2, j, hf);
    float t3[8];
    wmma4(aDt, bb, t3);                        // W1 = D16^T * Yq^T
    lds_transpose_b(lds[wv], t3, col, hf, bb);
    wmma4(aDt, bb, rec[c]);                    // rec = D16^T*Yq*D16 = IDCT*Yq*DCT
  }

  // YCbCr->RGB, clip, noise, shift, gamma, store inp.
#pragma unroll
  for (int v = 0; v < 8; ++v) {
    const float yv = rec[0][v];
    const float cb = rec[1][v] - 0.5f;
    const float cr = rec[2][v] - 0.5f;
    float rgb[3];
    rgb[0] = yv + 1.402f * cr;
    rgb[1] = yv - 0.34414f * cb - 0.71414f * cr;
    rgb[2] = yv + 1.772f * cb;
#pragma unroll
    for (int c = 0; c < 3; ++c) {
      const int off = c * PLANE + v * 512;
      float val = clip01(rgb[c]);
      const float gv = __builtin_nontemporal_load(gau_b + off);
      val = clip01(val + gv * 0.03f);
      val += sh[c];
      val = fminf(fmaxf(val, 1e-8f), 1.0f);
      __builtin_nontemporal_store(clip01(pow09(val)), outI + off);
    }
  }
}

extern "C" void kernel_launch(void* const* d_in, const int* in_sizes, int n_in,
                              void* d_out, int out_size, void* d_ws, size_t ws_size,
                              hipStream_t stream) {
  (void)in_sizes; (void)n_in; (void)out_size; (void)d_ws; (void)ws_size;
  const float* x      = (const float*)d_in[0];
  const float* dither = (const float*)d_in[1];
  const float* gauss  = (const float*)d_in[2];
  const float* shift  = (const float*)d_in[3];
  float* out = (float*)d_out;
  // 16384 waves (one 16x16x3 tile each) = 2048 blocks of 256 threads (8 waves)
  hipLaunchKernelGGL(bake_aug_kernel, dim3(2048), dim3(256), 0, stream,
                     x, dither, gauss, shift, out);
}